// SAGCN_8169027797547
// MI455X (gfx1250) — compile-verified
//
#include <hip/hip_runtime.h>

#define FEAT 40
#define BATCH 64
#define NPER0 2000
#define EDGES 2048000
#define NODES0 128000
#define KP1 1000
#define KP2 500
#define KP3 250
#define NCLS 91

typedef float v2f __attribute__((ext_vector_type(2)));
typedef float v8f __attribute__((ext_vector_type(8)));

// ---------------------------------------------------------------------------
// Generic fp32 WMMA GEMM: out[M,N] = A[M,K] @ W[N,K]^T (+bias) (optional relu)
// One wave computes one 16x16 output tile via V_WMMA_F32_16X16X4_F32.
// A layout (32-bit A, ISA 7.12.2): lane l supplies row M=l%16, VGPR j holds
// K = (l/16)*2 + j.  B mirrors A with N in place of M.  C/D: VGPR v holds
// M = v (+8 for lanes 16-31), N = l%16.
//
// Out-of-range M/N rows are CLAMPED (not predicated): a lane's A row feeds
// only output row M=lane%16 and a W row only output column N=lane%16, so
// clamped duplicates pollute only rows/cols masked at the store. This keeps
// the hot loop free of exec-mask branching. The K tail (K%4!=0, mlp2 and mlp3)
// uses branchless v_cndmask zeroing.
// ---------------------------------------------------------------------------
__global__ void gemm_wmma_f32(const float* __restrict__ A,
                              const float* __restrict__ W,
                              const float* __restrict__ bias,
                              float* __restrict__ out,
                              int M, int N, int K, int relu)
{
    int wave = (blockIdx.x * blockDim.x + threadIdx.x) >> 5;
    int lane = threadIdx.x & 31;
    int mtiles = (M + 15) >> 4;
    int ntiles = (N + 15) >> 4;
    if (wave >= mtiles * ntiles) return;          // wave-uniform exit
    int mt = wave / ntiles;
    int nt = wave % ntiles;

    int mrow = mt * 16 + (lane & 15);
    int nrow = nt * 16 + (lane & 15);
    int mrc = (mrow < M) ? mrow : (M - 1);        // clamped, address-safe
    int nrc = (nrow < N) ? nrow : (N - 1);
    int khalf = (lane >> 4) * 2;                  // lane-dependent K base

    const float* __restrict__ arow = A + (size_t)mrc * K + khalf;
    const float* __restrict__ brow = W + (size_t)nrc * K + khalf;

    v8f acc = {0.f, 0.f, 0.f, 0.f, 0.f, 0.f, 0.f, 0.f};
    int K4 = K & ~3;
    int k0 = 0;
    for (; k0 < K4; k0 += 4) {                    // unguarded hot loop
        v2f a, b;
        a[0] = arow[k0];
        a[1] = arow[k0 + 1];
        b[0] = brow[k0];
        b[1] = brow[k0 + 1];
        acc = __builtin_amdgcn_wmma_f32_16x16x4_f32(
            false, a, false, b, (short)0, acc, false, false);
    }
    if (k0 < K) {                                 // branchless K tail
        v2f a, b;
#pragma unroll
        for (int j = 0; j < 2; ++j) {
            int k = k0 + khalf + j;
            int kc = (k < K) ? k : (K - 1);
            float av = A[(size_t)mrc * K + kc];
            float bv = W[(size_t)nrc * K + kc];
            a[j] = (k < K) ? av : 0.0f;
            b[j] = (k < K) ? bv : 0.0f;
        }
        acc = __builtin_amdgcn_wmma_f32_16x16x4_f32(
            false, a, false, b, (short)0, acc, false, false);
    }

    int ncol = nt * 16 + (lane & 15);
#pragma unroll
    for (int v = 0; v < 8; ++v) {
        int m = mt * 16 + ((lane < 16) ? v : v + 8);
        if (m < M && ncol < N) {
            float r = acc[v] + (bias ? bias[ncol] : 0.0f);
            if (relu) r = fmaxf(r, 0.0f);
            out[m * N + ncol] = r;
        }
    }
}

// ---------------------------------------------------------------------------
// Edge / node utility kernels
// ---------------------------------------------------------------------------
__global__ void edge_init_kernel(const int* __restrict__ ei,
                                 const float* __restrict__ ea,
                                 int* row, int* col, float* mask, float* attr)
{
    int i = blockIdx.x * blockDim.x + threadIdx.x;
    if (i >= EDGES) return;
    row[i]  = ei[i];
    col[i]  = ei[EDGES + i];
    mask[i] = 1.0f;
    attr[i] = ea[i];
}

__global__ void idx_init_kernel(int* idxs)
{
    int i = blockIdx.x * blockDim.x + threadIdx.x;
    if (i >= NODES0) return;
    idxs[i] = i % NPER0;
}

__global__ void fill_f32_kernel(float* p, float v, int n)
{
    int i = blockIdx.x * blockDim.x + threadIdx.x;
    if (i < n) p[i] = v;
}

__global__ void fill_i32_kernel(int* p, int v, int n)
{
    int i = blockIdx.x * blockDim.x + threadIdx.x;
    if (i < n) p[i] = v;
}

__global__ void deg_scatter_kernel(const int* __restrict__ col,
                                   const float* __restrict__ attr,
                                   float* __restrict__ deg, int e)
{
    int i = blockIdx.x * blockDim.x + threadIdx.x;
    if (i >= e) return;
    float w = attr[i];
    if (w != 0.0f) atomicAdd(&deg[col[i]], w);
}

__global__ void rsqrt_kernel(float* deg, int n)
{
    int i = blockIdx.x * blockDim.x + threadIdx.x;
    if (i < n) deg[i] = rsqrtf(deg[i]);
}

// out = xw * dinv^2 + bias   (self-loop term of GCN)
__global__ void gcn_self_kernel(const float* __restrict__ xw,
                                const float* __restrict__ dinv,
                                const float* __restrict__ bias,
                                float* __restrict__ out, int n)
{
    int idx = blockIdx.x * blockDim.x + threadIdx.x;
    if (idx >= n * FEAT) return;
    int i = idx / FEAT, f = idx % FEAT;
    float d = dinv[i];
    out[idx] = xw[idx] * d * d + bias[f];
}

// weighted message scatter: out[col] += dinv[row]*w*dinv[col] * xw[row]
// float4 gathers (global_load_b128): xw rows are 160B and 16B-aligned.
__global__ void msg_scatter_kernel(const int* __restrict__ row,
                                   const int* __restrict__ col,
                                   const float* __restrict__ attr,
                                   const float* __restrict__ dinv,
                                   const float* __restrict__ xw,
                                   float* __restrict__ out, int e)
{
    int i = blockIdx.x * blockDim.x + threadIdx.x;
    if (i >= e) return;
    float w = attr[i];
    if (w == 0.0f) return;
    int r = row[i], c = col[i];
    float cf = dinv[r] * w * dinv[c];
    const float4* __restrict__ src4 = (const float4*)(xw + (size_t)r * FEAT);
    float* __restrict__ dst = out + (size_t)c * FEAT;
#pragma unroll
    for (int q = 0; q < FEAT / 4; ++q) {
        float4 s = src4[q];
        atomicAdd(&dst[4 * q + 0], cf * s.x);
        atomicAdd(&dst[4 * q + 1], cf * s.y);
        atomicAdd(&dst[4 * q + 2], cf * s.z);
        atomicAdd(&dst[4 * q + 3], cf * s.w);
    }
}

__global__ void relu_kernel(float* x, int n)
{
    int i = blockIdx.x * blockDim.x + threadIdx.x;
    if (i < n) x[i] = fmaxf(x[i], 0.0f);
}

// scorer aggregation: aggr[col] += mask * x[row]
__global__ void aggr_scatter_kernel(const int* __restrict__ row,
                                    const int* __restrict__ col,
                                    const float* __restrict__ mask,
                                    const float* __restrict__ x,
                                    float* __restrict__ aggr, int e)
{
    int i = blockIdx.x * blockDim.x + threadIdx.x;
    if (i >= e) return;
    if (mask[i] <= 0.0f) return;
    int r = row[i], c = col[i];
    const float4* __restrict__ src4 = (const float4*)(x + (size_t)r * FEAT);
    float* __restrict__ dst = aggr + (size_t)c * FEAT;
#pragma unroll
    for (int q = 0; q < FEAT / 4; ++q) {
        float4 s = src4[q];
        atomicAdd(&dst[4 * q + 0], s.x);
        atomicAdd(&dst[4 * q + 1], s.y);
        atomicAdd(&dst[4 * q + 2], s.z);
        atomicAdd(&dst[4 * q + 3], s.w);
    }
}

__global__ void score_kernel(const float* __restrict__ x,
                             const float* __restrict__ aggr,
                             const float* __restrict__ relw,
                             const float* __restrict__ relb,
                             const float* __restrict__ rootw,
                             float* __restrict__ score, int n)
{
    int i = blockIdx.x * blockDim.x + threadIdx.x;
    if (i >= n) return;
    float s = relb[0];
#pragma unroll 8
    for (int f = 0; f < FEAT; ++f)
        s += aggr[i * FEAT + f] * relw[f] + x[i * FEAT + f] * rootw[f];
    score[i] = tanhf(s);
}

// per-graph iterative top-k (descending, lower-index tie-break == lax.top_k)
__global__ void topk_kernel(const float* __restrict__ score,
                            int* __restrict__ perm, int n_per, int k)
{
    __shared__ float sv[2048];
    __shared__ float rv[256];
    __shared__ int   ri[256];
    int g = blockIdx.x;
    const float* s = score + g * n_per;
    for (int i = threadIdx.x; i < n_per; i += blockDim.x) sv[i] = s[i];
    __syncthreads();
    for (int j = 0; j < k; ++j) {
        float best = -1e30f;
        int bi = n_per;
        for (int i = threadIdx.x; i < n_per; i += blockDim.x) {
            float v = sv[i];
            if (v > best) { best = v; bi = i; }
        }
        rv[threadIdx.x] = best;
        ri[threadIdx.x] = bi;
        __syncthreads();
        for (int off = blockDim.x >> 1; off > 0; off >>= 1) {
            if (threadIdx.x < off) {
                float v2 = rv[threadIdx.x + off];
                int   i2 = ri[threadIdx.x + off];
                if (v2 > rv[threadIdx.x] ||
                    (v2 == rv[threadIdx.x] && i2 < ri[threadIdx.x])) {
                    rv[threadIdx.x] = v2;
                    ri[threadIdx.x] = i2;
                }
            }
            __syncthreads();
        }
        if (threadIdx.x == 0) {
            perm[g * k + j] = g * n_per + ri[0];
            sv[ri[0]] = -1e30f;
        }
        __syncthreads();
    }
}

__global__ void nodemap_set_kernel(const int* __restrict__ perm,
                                   int* __restrict__ nodemap, int m)
{
    int j = blockIdx.x * blockDim.x + threadIdx.x;
    if (j < m) nodemap[perm[j]] = j;
}

__global__ void gather_x_kernel(const float* __restrict__ x,
                                const float* __restrict__ score,
                                const int* __restrict__ perm,
                                float* __restrict__ xnew, int m)
{
    int idx = blockIdx.x * blockDim.x + threadIdx.x;
    if (idx >= m * FEAT) return;
    int j = idx / FEAT, f = idx % FEAT;
    int p = perm[j];
    xnew[idx] = x[p * FEAT + f] * score[p];
}

__global__ void gather_idx_kernel(const int* __restrict__ idxs,
                                  const int* __restrict__ perm,
                                  int* __restrict__ idxs_new, int m)
{
    int j = blockIdx.x * blockDim.x + threadIdx.x;
    if (j < m) idxs_new[j] = idxs[perm[j]];
}

__global__ void remap_edges_kernel(int* row, int* col, float* mask, float* attr,
                                   const int* __restrict__ nodemap, int e)
{
    int i = blockIdx.x * blockDim.x + threadIdx.x;
    if (i >= e) return;
    int nr = nodemap[row[i]];
    int nc = nodemap[col[i]];
    bool keep = (nr >= 0) && (nc >= 0) && (mask[i] > 0.0f);
    float mf = keep ? 1.0f : 0.0f;
    mask[i] = mf;
    attr[i] *= mf;
    row[i] = keep ? nr : 0;
    col[i] = keep ? nc : 0;
}

// per-node mean/max readout -> [B, 2*K3]
__global__ void readout_kernel(const float* __restrict__ x,
                               float* __restrict__ ro, int n)
{
    int i = blockIdx.x * blockDim.x + threadIdx.x;
    if (i >= n) return;
    float s = 0.0f, m = -1e30f;
#pragma unroll 8
    for (int f = 0; f < FEAT; ++f) {
        float v = x[i * FEAT + f];
        s += v;
        m = fmaxf(m, v);
    }
    int b = i / KP3, c = i % KP3;
    ro[b * (2 * KP3) + c]        = s * (1.0f / FEAT);
    ro[b * (2 * KP3) + KP3 + c]  = m;
}

__global__ void conv1d_kernel(const float* __restrict__ ro,
                              const float* __restrict__ w,
                              const float* __restrict__ bias,
                              float* __restrict__ out)
{
    int idx = blockIdx.x * blockDim.x + threadIdx.x;
    if (idx >= BATCH * 2 * KP3) return;
    int b = idx / (2 * KP3), i = idx % (2 * KP3);
    float acc = bias[0];
#pragma unroll
    for (int t = 0; t < 5; ++t) {
        int p = i + t - 2;
        if (p >= 0 && p < 2 * KP3) acc += ro[b * (2 * KP3) + p] * w[t];
    }
    out[idx] = acc;
}

__global__ void log_softmax_kernel(const float* __restrict__ in,
                                   float* __restrict__ out, int C)
{
    __shared__ float red[128];
    int b = blockIdx.x;
    const float* x = in + b * C;
    float m = -1e30f;
    for (int i = threadIdx.x; i < C; i += blockDim.x) m = fmaxf(m, x[i]);
    red[threadIdx.x] = m;
    __syncthreads();
    for (int off = blockDim.x >> 1; off > 0; off >>= 1) {
        if (threadIdx.x < off)
            red[threadIdx.x] = fmaxf(red[threadIdx.x], red[threadIdx.x + off]);
        __syncthreads();
    }
    m = red[0];
    __syncthreads();
    float s = 0.0f;
    for (int i = threadIdx.x; i < C; i += blockDim.x) s += expf(x[i] - m);
    red[threadIdx.x] = s;
    __syncthreads();
    for (int off = blockDim.x >> 1; off > 0; off >>= 1) {
        if (threadIdx.x < off) red[threadIdx.x] += red[threadIdx.x + off];
        __syncthreads();
    }
    float lse = m + logf(red[0]);
    for (int i = threadIdx.x; i < C; i += blockDim.x) out[b * C + i] = x[i] - lse;
}

__global__ void idx_to_float_kernel(const int* __restrict__ idxs,
                                    float* __restrict__ out, int n)
{
    int i = blockIdx.x * blockDim.x + threadIdx.x;
    if (i < n) out[i] = (float)idxs[i];
}

// ---------------------------------------------------------------------------
// Host orchestration
// ---------------------------------------------------------------------------
extern "C" void kernel_launch(void* const* d_in, const int* in_sizes, int n_in,
                              void* d_out, int out_size, void* d_ws, size_t ws_size,
                              hipStream_t stream)
{
    const float* d_x       = (const float*)d_in[0];
    const int*   d_ei      = (const int*)d_in[1];
    const float* d_ea      = (const float*)d_in[2];
    const float* conv_w[3] = {(const float*)d_in[3], (const float*)d_in[5], (const float*)d_in[7]};
    const float* conv_b[3] = {(const float*)d_in[4], (const float*)d_in[6], (const float*)d_in[8]};
    const float* relw[3]   = {(const float*)d_in[9],  (const float*)d_in[12], (const float*)d_in[15]};
    const float* relb[3]   = {(const float*)d_in[10], (const float*)d_in[13], (const float*)d_in[16]};
    const float* rootw[3]  = {(const float*)d_in[11], (const float*)d_in[14], (const float*)d_in[17]};
    const float* cnn_w  = (const float*)d_in[18];
    const float* cnn_b  = (const float*)d_in[19];
    const float* mlp1_w = (const float*)d_in[20];
    const float* mlp1_b = (const float*)d_in[21];
    const float* mlp2_w = (const float*)d_in[22];
    const float* mlp2_b = (const float*)d_in[23];
    const float* mlp3_w = (const float*)d_in[24];
    const float* mlp3_b = (const float*)d_in[25];
    float* out = (float*)d_out;

    // bump allocator on workspace
    char* wsp = (char*)d_ws;
    auto alloc = [&](size_t bytes) -> void* {
        void* p = (void*)wsp;
        wsp += (bytes + 255) & ~(size_t)255;
        return p;
    };
    int*   rowb    = (int*)alloc(EDGES * 4);
    int*   colb    = (int*)alloc(EDGES * 4);
    float* maskb   = (float*)alloc(EDGES * 4);
    float* attrb   = (float*)alloc(EDGES * 4);
    float* xA      = (float*)alloc((size_t)BATCH * KP1 * FEAT * 4);   // pooled x
    float* xB      = (float*)alloc((size_t)NODES0 * FEAT * 4);        // conv out
    float* xw      = (float*)alloc((size_t)NODES0 * FEAT * 4);        // xw / aggr (aliased)
    float* aggr    = xw;
    float* deg     = (float*)alloc(NODES0 * 4);
    float* score   = (float*)alloc(NODES0 * 4);
    int*   perm    = (int*)alloc(BATCH * KP1 * 4);
    int*   nodemap = (int*)alloc(NODES0 * 4);
    int*   idxsA   = (int*)alloc(NODES0 * 4);
    int*   idxsB   = (int*)alloc(BATCH * KP1 * 4);
    float* readout = (float*)alloc(BATCH * 2 * KP3 * 4);
    float* convo   = (float*)alloc(BATCH * 2 * KP3 * 4);
    float* h1      = (float*)alloc(BATCH * KP3 * 4);
    float* h2      = (float*)alloc(BATCH * (KP3 / 2) * 4);
    float* logits  = (float*)alloc(BATCH * NCLS * 4);

    auto cdiv = [](int a, int b) { return (a + b - 1) / b; };
    const int TB = 256;

    auto gemm = [&](const float* A, const float* W, const float* bias,
                    float* o, int M, int N, int K, int relu) {
        int waves = cdiv(M, 16) * cdiv(N, 16);
        int threads = waves * 32;
        gemm_wmma_f32<<<cdiv(threads, 128), 128, 0, stream>>>(A, W, bias, o, M, N, K, relu);
    };

    auto run_gcn = [&](const float* xin, int n, const float* W, const float* b,
                       float* o) {
        fill_f32_kernel<<<cdiv(n, TB), TB, 0, stream>>>(deg, 1.0f, n);
        deg_scatter_kernel<<<cdiv(EDGES, TB), TB, 0, stream>>>(colb, attrb, deg, EDGES);
        rsqrt_kernel<<<cdiv(n, TB), TB, 0, stream>>>(deg, n);   // deg -> dinv
        gemm(xin, W, nullptr, xw, n, FEAT, FEAT, 0);
        gcn_self_kernel<<<cdiv(n * FEAT, TB), TB, 0, stream>>>(xw, deg, b, o, n);
        msg_scatter_kernel<<<cdiv(EDGES, TB), TB, 0, stream>>>(rowb, colb, attrb, deg, xw, o, EDGES);
        relu_kernel<<<cdiv(n * FEAT, TB), TB, 0, stream>>>(o, n * FEAT);
    };

    int* idx_cur = idxsA;
    int* idx_nxt = idxsB;
    auto run_pool = [&](const float* xin, int n, int n_per, int k, int stage,
                        float* xnew) {
        int m = BATCH * k;
        fill_f32_kernel<<<cdiv(n * FEAT, TB), TB, 0, stream>>>(aggr, 0.0f, n * FEAT);
        aggr_scatter_kernel<<<cdiv(EDGES, TB), TB, 0, stream>>>(rowb, colb, maskb, xin, aggr, EDGES);
        score_kernel<<<cdiv(n, TB), TB, 0, stream>>>(xin, aggr, relw[stage], relb[stage],
                                                     rootw[stage], score, n);
        topk_kernel<<<BATCH, 256, 0, stream>>>(score, perm, n_per, k);
        fill_i32_kernel<<<cdiv(n, TB), TB, 0, stream>>>(nodemap, -1, n);
        nodemap_set_kernel<<<cdiv(m, TB), TB, 0, stream>>>(perm, nodemap, m);
        gather_x_kernel<<<cdiv(m * FEAT, TB), TB, 0, stream>>>(xin, score, perm, xnew, m);
        gather_idx_kernel<<<cdiv(m, TB), TB, 0, stream>>>(idx_cur, perm, idx_nxt, m);
        remap_edges_kernel<<<cdiv(EDGES, TB), TB, 0, stream>>>(rowb, colb, maskb, attrb,
                                                               nodemap, EDGES);
        int* t = idx_cur; idx_cur = idx_nxt; idx_nxt = t;   // host ptr swap (deterministic)
    };

    // ---- init mutable edge state & idxs ----
    edge_init_kernel<<<cdiv(EDGES, TB), TB, 0, stream>>>(d_ei, d_ea, rowb, colb, maskb, attrb);
    idx_init_kernel<<<cdiv(NODES0, TB), TB, 0, stream>>>(idxsA);

    // ---- stage 1 ----
    run_gcn(d_x, NODES0, conv_w[0], conv_b[0], xB);
    run_pool(xB, NODES0, NPER0, KP1, 0, xA);
    // ---- stage 2 ----
    run_gcn(xA, BATCH * KP1, conv_w[1], conv_b[1], xB);
    run_pool(xB, BATCH * KP1, KP1, KP2, 1, xA);
    // ---- stage 3 ----
    run_gcn(xA, BATCH * KP2, conv_w[2], conv_b[2], xB);
    run_pool(xB, BATCH * KP2, KP2, KP3, 2, xA);

    // ---- readout / head ----
    int nfin = BATCH * KP3;
    readout_kernel<<<cdiv(nfin, TB), TB, 0, stream>>>(xA, readout, nfin);
    conv1d_kernel<<<cdiv(BATCH * 2 * KP3, TB), TB, 0, stream>>>(readout, cnn_w, cnn_b, convo);
    gemm(convo, mlp1_w, mlp1_b, h1, BATCH, KP3, 2 * KP3, 1);
    gemm(h1, mlp2_w, mlp2_b, h2, BATCH, KP3 / 2, KP3, 1);
    gemm(h2, mlp3_w, mlp3_b, logits, BATCH, NCLS, KP3 / 2, 0);
    log_softmax_kernel<<<BATCH, 128, 0, stream>>>(logits, out, NCLS);
    idx_to_float_kernel<<<cdiv(nfin, TB), TB, 0, stream>>>(idx_cur, out + BATCH * NCLS, nfin);
}